// Len_Trim_Avg_77403900608622
// MI455X (gfx1250) — compile-verified
//
#include <hip/hip_runtime.h>
#include <cstdint>
#include <cstddef>

// ---------------- problem constants (match reference) ----------------
#define B_DIM    32
#define T_DIM    4096
#define D_DIM    256
#define DT       16            // features per workgroup
#define NTHREADS 256
#define NWAVES   8
#define QV       0.1f

// ---------------- LDS layout (dwords) ----------------
#define HIST_OFF      0
#define HIST_PER_WAVE (4 * 257)                 // 4 bank-offset replicas (pass 0) / lo+hi hists (passes 1-3)
#define HIST_DW       (NWAVES * HIST_PER_WAVE)  // 8224
#define KEYS_OFF      (HIST_OFF + HIST_DW)      // 8224 (dword-multiple of 4 -> 16B aligned)
#define KEYS_STRIDE   17                        // pad 16 -> 17 for conflict-free column scans
#define KEYS_DW       (T_DIM * KEYS_STRIDE)     // 69632
#define LDS_DW        (KEYS_OFF + KEYS_DW)      // 77856
#define LDS_BYTES     (LDS_DW * 4)              // 311424  (< 320KB WGP LDS)

// ---------------- CDNA5 async global->LDS path ----------------
#ifndef __has_builtin
#define __has_builtin(x) 0
#endif

#if __has_builtin(__builtin_amdgcn_global_load_async_to_lds_b128)
#define HAVE_ASYNC_LDS 1
#else
#define HAVE_ASYNC_LDS 0
#endif

#if HAVE_ASYNC_LDS
// 16-byte int vector in global (as1) / LDS (as3) address spaces, matching the
// builtin's parameter types reported by clang.
typedef int v4i __attribute__((vector_size(16)));
typedef __attribute__((address_space(1))) v4i gv4i;
typedef __attribute__((address_space(3))) v4i lv4i;
#endif

#if __has_builtin(__builtin_amdgcn_s_wait_asynccnt)
#define WAIT_ASYNC(n) __builtin_amdgcn_s_wait_asynccnt(n)
#elif defined(__AMDGCN__)
#define WAIT_ASYNC(n) asm volatile("s_wait_asynccnt %0" ::"i"(n) : "memory")
#else
#define WAIT_ASYNC(n) do { } while (0)
#endif

// float -> order-preserving u32 key (and back)
__device__ __forceinline__ unsigned f2k(unsigned u) {
  return u ^ ((unsigned)((int)u >> 31) | 0x80000000u);
}
__device__ __forceinline__ float k2f(unsigned k) {
  unsigned m = ((int)k < 0) ? 0x80000000u : 0xFFFFFFFFu;
  return __uint_as_float(k ^ m);
}

__device__ __forceinline__ unsigned waveInclScan(unsigned v, int lane) {
#pragma unroll
  for (int d = 1; d < 32; d <<= 1) {
    unsigned o = __shfl_up(v, (unsigned)d, 32);
    if (lane >= d) v += o;
  }
  return v;
}

// Locate which 8-bit digit contains rank `need` among current candidates.
// cbin[0..7] are this lane's bin counts (bins lane*8 .. lane*8+7),
// excl/incl are the exclusive/inclusive wave-scan of this lane's total.
__device__ __forceinline__ void selectDigit(const unsigned cbin[8], unsigned excl,
                                            unsigned incl, int lane, int& need,
                                            unsigned& pref, int shift) {
  bool found = ((unsigned)need >= excl) && ((unsigned)need < incl);
  int dig = 0, bef = 0;
  if (found) {
    unsigned acc = excl;
#pragma unroll
    for (int j = 0; j < 8; ++j) {
      if ((unsigned)need < acc + cbin[j]) { dig = lane * 8 + j; bef = (int)acc; break; }
      acc += cbin[j];
    }
  }
  unsigned long long bal = __ballot(found);
  int src = __ffsll(bal) - 1;
  dig = __shfl(dig, src, 32);
  bef = __shfl(bef, src, 32);
  need -= bef;
  pref |= ((unsigned)dig) << shift;
}

__global__ __launch_bounds__(NTHREADS, 1)
void trimmed_mean_kernel(const float* __restrict__ outputs,
                         const float* __restrict__ lens,
                         float* __restrict__ out) {
  extern __shared__ unsigned lds[];
  const int tid  = threadIdx.x;
  const int lane = tid & 31;
  const int wave = tid >> 5;
  const int b    = blockIdx.y;
  const int d0   = blockIdx.x * DT;

  // ---- valid length & target ranks (replicate reference f32 math; RNE round) ----
  const float length = lens[b] * (float)T_DIM;
  int L = (length == 1.0f) ? T_DIM : ((int)floorf(length) + 1);
  L = (L < T_DIM) ? L : T_DIM;
  const float Lf = (float)L;
  const int kmin = (int)rintf(QV * (Lf - 1.0f));
  const int kmax = (int)rintf((1.0f - QV) * (Lf - 1.0f));

  // =================== Phase 1: async DMA tile into LDS ===================
  // Raw floats land row-major [t][f] at KEYS_OFF; rows t >= L are skipped
  // (they are replaced by +inf-equivalent pad keys during conversion).
  {
    const float* gsrc = outputs + (size_t)b * T_DIM * D_DIM + d0;
#pragma unroll 1
    for (int u = 0; u < (T_DIM * 4) / NTHREADS; ++u) {  // 64 x (b128 per thread)
      const int unit = u * NTHREADS + tid;              // (row, 16B-quad)
      const int row  = unit >> 2;
      const int qq   = (unit & 3) * 4;
      if (row < L) {
        const float* gp = gsrc + (size_t)row * D_DIM + qq;
        unsigned* lp = &lds[KEYS_OFF + row * 16 + qq];
#if HAVE_ASYNC_LDS
        __builtin_amdgcn_global_load_async_to_lds_b128((gv4i*)gp, (lv4i*)lp, 0, 0);
#else
        *(float4*)lp = *(const float4*)gp;
#endif
      }
    }
  }
  WAIT_ASYNC(0);
  __syncthreads();

  // ====== Phase 2: in-place stride-16 -> stride-17 expand + key convert ======
  // Process 256-row chunks in DESCENDING order: chunk cc writes dwords
  // [4352*cc, 4352*cc+4352) which never clobbers unread source dwords
  // (< 4096*cc) of lower chunks; same-chunk sources are read to registers
  // before the barrier that precedes the writes.
#pragma unroll 1
  for (int cc = (T_DIM / NTHREADS) - 1; cc >= 0; --cc) {
    const int t = cc * NTHREADS + tid;
    uint4 r0 = *(const uint4*)&lds[KEYS_OFF + t * 16 + 0];
    uint4 r1 = *(const uint4*)&lds[KEYS_OFF + t * 16 + 4];
    uint4 r2 = *(const uint4*)&lds[KEYS_OFF + t * 16 + 8];
    uint4 r3 = *(const uint4*)&lds[KEYS_OFF + t * 16 + 12];
    __syncthreads();
    unsigned v[16] = {r0.x, r0.y, r0.z, r0.w, r1.x, r1.y, r1.z, r1.w,
                      r2.x, r2.y, r2.z, r2.w, r3.x, r3.y, r3.z, r3.w};
    const bool valid = (t < L);
#pragma unroll
    for (int j = 0; j < 16; ++j) {
      lds[KEYS_OFF + t * KEYS_STRIDE + j] = valid ? f2k(v[j]) : 0xFFFFFFFFu;
    }
    __syncthreads();
  }

  // =================== Phase 3: per-wave dual radix-select ===================
  // Wave w handles columns f = w and f = w + 8. All histogram state is
  // per-wave (LDS same-wave ordering is guaranteed), so no barriers needed.
  unsigned* wh = &lds[HIST_OFF + wave * HIST_PER_WAVE];
  const int lane8 = lane * 8;

#pragma unroll 1
  for (int rnd = 0; rnd < 2; ++rnd) {
    const int f = wave + NWAVES * rnd;
    unsigned prefLo = 0u, prefHi = 0u;
    int needLo = kmin, needHi = kmax;

#pragma unroll 1
    for (int p = 0; p < 4; ++p) {
      const int shift = 24 - 8 * p;
      for (int j = lane; j < HIST_PER_WAVE; j += 32) wh[j] = 0u;

      if (p == 0) {
        // Shared histogram for both ranks; 4 bank-offset replicas (stride 257)
        // spread hot exponent bins across banks.
        const int rep = (lane & 3) * 257;
#pragma unroll 4
        for (int i = 0; i < T_DIM / 32; ++i) {
          unsigned k = lds[KEYS_OFF + (lane + 32 * i) * KEYS_STRIDE + f];
          atomicAdd(&wh[rep + (k >> 24)], 1u);
        }
      } else {
        const unsigned maskA = 0xFFFFFFFFu << (shift + 8);
#pragma unroll 4
        for (int i = 0; i < T_DIM / 32; ++i) {
          unsigned k = lds[KEYS_OFF + (lane + 32 * i) * KEYS_STRIDE + f];
          const unsigned dig = (k >> shift) & 255u;
          if ((k & maskA) == prefLo) atomicAdd(&wh[dig], 1u);
          if ((k & maskA) == prefHi) atomicAdd(&wh[514 + dig], 1u);
        }
      }

      if (p == 0) {
        unsigned cbin[8], tot = 0u;
#pragma unroll
        for (int j = 0; j < 8; ++j) {
          const int bb = lane8 + j;
          cbin[j] = wh[bb] + wh[bb + 257] + wh[bb + 514] + wh[bb + 771];
          tot += cbin[j];
        }
        unsigned incl = waveInclScan(tot, lane);
        selectDigit(cbin, incl - tot, incl, lane, needLo, prefLo, shift);
        selectDigit(cbin, incl - tot, incl, lane, needHi, prefHi, shift);
      } else {
        {
          unsigned cbin[8], tot = 0u;
#pragma unroll
          for (int j = 0; j < 8; ++j) { cbin[j] = wh[lane8 + j]; tot += cbin[j]; }
          unsigned incl = waveInclScan(tot, lane);
          selectDigit(cbin, incl - tot, incl, lane, needLo, prefLo, shift);
        }
        {
          unsigned cbin[8], tot = 0u;
#pragma unroll
          for (int j = 0; j < 8; ++j) { cbin[j] = wh[514 + lane8 + j]; tot += cbin[j]; }
          unsigned incl = waveInclScan(tot, lane);
          selectDigit(cbin, incl - tot, incl, lane, needHi, prefHi, shift);
        }
      }
    }

    // ---- Phase 4: trimmed sum over [a_min, a_max] (key order == float order) ----
    const unsigned keyLo = prefLo, keyHi = prefHi;
    float s = 0.0f;
    int cnt = 0;
#pragma unroll 4
    for (int i = 0; i < T_DIM / 32; ++i) {
      const int t = lane + 32 * i;
      const unsigned k = lds[KEYS_OFF + t * KEYS_STRIDE + f];
      if (t < L && k >= keyLo && k <= keyHi) {
        s += k2f(k);
        ++cnt;
      }
    }
#pragma unroll
    for (int o = 16; o > 0; o >>= 1) {
      s += __shfl_xor(s, o, 32);
      cnt += __shfl_xor(cnt, o, 32);
    }
    if (lane == 0) out[b * D_DIM + d0 + f] = s / (float)cnt;  // cnt = L - n_max - n_min
  }
}

extern "C" void kernel_launch(void* const* d_in, const int* in_sizes, int n_in,
                              void* d_out, int out_size, void* d_ws, size_t ws_size,
                              hipStream_t stream) {
  (void)in_sizes; (void)n_in; (void)out_size; (void)d_ws; (void)ws_size;
  const float* outputs = (const float*)d_in[0];
  const float* lens    = (const float*)d_in[1];
  float* out           = (float*)d_out;

  static_assert(LDS_BYTES <= 320 * 1024, "exceeds CDNA5 WGP LDS");
  (void)hipFuncSetAttribute(reinterpret_cast<const void*>(&trimmed_mean_kernel),
                            hipFuncAttributeMaxDynamicSharedMemorySize, LDS_BYTES);

  dim3 grid(D_DIM / DT, B_DIM);   // 16 x 32 = 512 workgroups
  trimmed_mean_kernel<<<grid, NTHREADS, LDS_BYTES, stream>>>(outputs, lens, out);
}